// AdaPool2d_309237645644
// MI455X (gfx1250) — compile-verified
//
#include <hip/hip_runtime.h>
#include <stdint.h>

// ---------------------------------------------------------------------------
// AdaPool2d (K=stride=2) for MI455X / gfx1250.
// Memory-bound streaming kernel: TDM (tensor_load_to_lds) double-buffered
// input staging, LDS compute, non-temporal output stores.
// ---------------------------------------------------------------------------

typedef unsigned int v4u __attribute__((ext_vector_type(4)));
typedef int          v8i __attribute__((ext_vector_type(8)));
typedef int          v4i __attribute__((ext_vector_type(4)));

#define IN_W        224
#define IN_H        224
#define OUT_W       112
#define CHUNK_ROWS  32                      // input rows per TDM chunk
#define CHUNK_ELTS  (IN_W * CHUNK_ROWS)     // 7168 floats = 28 KB
#define NCHUNK      (IN_H / CHUNK_ROWS)     // 7
#define OROWS_PER_CHUNK (CHUNK_ROWS / 2)    // 16 output rows
#define OUTS_PER_CHUNK  (OROWS_PER_CHUNK * OUT_W)  // 1792 = 7 * 256

#define F32_MAX 3.402823466e+38f

// Issue one TDM load: contiguous CHUNK_ELTS f32 from global -> LDS.
// Descriptor layout per CDNA5 ISA ch.8 (D# groups 0..3).
__device__ __forceinline__ void tdm_load_chunk(const float* gsrc, unsigned lds_byte_addr) {
    unsigned long long ga = (unsigned long long)(uintptr_t)gsrc;

    v4u g0;
    g0.x = 1u;                                    // count=1 (valid), user mode, no gather
    g0.y = lds_byte_addr;                         // lds_addr [63:32]
    g0.z = (unsigned)(ga & 0xFFFFFFFFu);          // global_addr lo
    g0.w = (unsigned)((ga >> 32) & 0x01FFFFFFu)   // global_addr [56:32]
         | 0x80000000u;                           // type=2 ("image") at [127:126]

    v8i g1;
    g1[0] = (int)(2u << 16);                          // wg_mask=0 (not in cluster), data_size=4B
    g1[1] = (int)((CHUNK_ELTS & 0xFFFFu) << 16);      // tensor_dim0 lo16 (bits 63:48->[31:16])
    g1[2] = (int)(((CHUNK_ELTS >> 16) & 0xFFFFu)      // tensor_dim0 hi16
         | (1u << 16));                               // tensor_dim1 = 1 (lo16)
    g1[3] = (int)(((unsigned)CHUNK_ELTS) << 16);      // tensor_dim1 hi16=0 | tile_dim0=7168
    g1[4] = 1;                                        // tile_dim1 = 1, tile_dim2 = 0
    g1[5] = CHUNK_ELTS;                               // tensor_dim0_stride lo32
    g1[6] = 0;                                        // stride0 hi16 | stride1 lo16
    g1[7] = 0;                                        // stride1 hi32

    v4i g2; g2[0] = 1; g2[1] = 1; g2[2] = 0; g2[3] = 0;  // tensor_dim2=1, tensor_dim3=1, tile_dim3=0
    v4i g3; g3[0] = 0; g3[1] = 0; g3[2] = 0; g3[3] = 0;  // stride3=0, tensor_dim4=0, tile_dim4=0

    // Extra 8-dword operand on this toolchain's 6-arg builtin (zero-filled,
    // matching the probe-verified call shape).
    v8i g4; g4[0] = 0; g4[1] = 0; g4[2] = 0; g4[3] = 0;
            g4[4] = 0; g4[5] = 0; g4[6] = 0; g4[7] = 0;

    __builtin_amdgcn_tensor_load_to_lds(g0, g1, g2, g3, g4, 0);
}

__device__ __forceinline__ float adapool4(float x0, float x1, float x2, float x3, float b) {
    // EDSCW: softmax over Dice coefficient vs window mean, weights the taps.
    float avg = 0.25f * ((x0 + x1) + (x2 + x3));
    float a2  = avg * avg;
    float d0 = (2.0f * x0 * avg) / (x0 * x0 + a2);
    float d1 = (2.0f * x1 * avg) / (x1 * x1 + a2);
    float d2 = (2.0f * x2 * avg) / (x2 * x2 + a2);
    float d3 = (2.0f * x3 * avg) / (x3 * x3 + a2);
    float md = fmaxf(fmaxf(d0, d1), fmaxf(d2, d3));
    float e0 = __expf(d0 - md), e1 = __expf(d1 - md);
    float e2 = __expf(d2 - md), e3 = __expf(d3 - md);
    float es = (e0 + e1) + (e2 + e3);
    float edscw = ((e0 * x0 + e1 * x1) + (e2 * x2 + e3 * x3)) / es;

    // EM: SoftPool (softmax over raw taps).
    float mx = fmaxf(fmaxf(x0, x1), fmaxf(x2, x3));
    float f0 = __expf(x0 - mx), f1 = __expf(x1 - mx);
    float f2 = __expf(x2 - mx), f3 = __expf(x3 - mx);
    float fs = (f0 + f1) + (f2 + f3);
    float em = ((f0 * x0 + f1 * x1) + (f2 * x2 + f3 * x3)) / fs;

    float r = b * edscw + (1.0f - b) * em;
    // jnp.nan_to_num: NaN -> 0, +/-inf -> +/-FLT_MAX
    if (__builtin_isnan(r)) r = 0.0f;
    r = fminf(fmaxf(r, -F32_MAX), F32_MAX);
    return r;
}

__global__ __launch_bounds__(256) void adapool2d_tdm_kernel(
        const float* __restrict__ x,       // [1024, 224, 224] (B*C planes)
        const float* __restrict__ beta,    // [112, 112]
        float* __restrict__ out) {         // [1024, 112, 112]
    __shared__ float lds[2][CHUNK_ELTS];   // 57,344 B double buffer

    const int plane = blockIdx.x;          // 0..1023
    const int tid   = threadIdx.x;
    const float* xp = x   + (size_t)plane * (IN_W * IN_H);
    float*       op = out + (size_t)plane * (OUT_W * OUT_W);

    // LDS byte offsets of the two staging buffers (generic ptr low 32 bits).
    const unsigned lbuf[2] = { (unsigned)(uintptr_t)&lds[0][0],
                               (unsigned)(uintptr_t)&lds[1][0] };

    const bool leader = (tid < 32);        // wave 0 drives the TDM pipeline

    if (leader) tdm_load_chunk(xp, lbuf[0]);

    for (int c = 0; c < NCHUNK; ++c) {
        const int cur = c & 1;
        if (leader) {
            if (c + 1 < NCHUNK) {
                tdm_load_chunk(xp + (size_t)(c + 1) * CHUNK_ELTS, lbuf[(c + 1) & 1]);
                __builtin_amdgcn_s_wait_tensorcnt(1);  // oldest (chunk c) done
            } else {
                __builtin_amdgcn_s_wait_tensorcnt(0);
            }
        }
        __syncthreads();                   // chunk c visible to all 8 waves
        asm volatile("" ::: "memory");     // LDS was written by TDM, not by threads

        const float* buf = &lds[cur][0];
        #pragma unroll
        for (int k = 0; k < OUTS_PER_CHUNK / 256; ++k) {   // 7 iterations
            const int i  = tid + k * 256;
            const int oy = i / OUT_W;                       // 0..15
            const int ox = i - oy * OUT_W;
            const float2 t0 = *(const float2*)(buf + (2 * oy)     * IN_W + 2 * ox);
            const float2 t1 = *(const float2*)(buf + (2 * oy + 1) * IN_W + 2 * ox);
            const int gy = c * OROWS_PER_CHUNK + oy;
            const float b = beta[gy * OUT_W + ox];
            const float r = adapool4(t0.x, t0.y, t1.x, t1.y, b);
            // Streamed once, never re-read: bypass-ish NT store.
            __builtin_nontemporal_store(r, op + gy * OUT_W + ox);
        }
        __syncthreads();                   // all reads of buf[cur] done before reuse
    }
}

extern "C" void kernel_launch(void* const* d_in, const int* in_sizes, int n_in,
                              void* d_out, int out_size, void* d_ws, size_t ws_size,
                              hipStream_t stream) {
    const float* x    = (const float*)d_in[0];   // 16*64*224*224 f32
    const float* beta = (const float*)d_in[1];   // 112*112 f32
    float*       out  = (float*)d_out;           // 16*64*112*112 f32
    (void)in_sizes; (void)n_in; (void)out_size; (void)d_ws; (void)ws_size;

    dim3 grid(16 * 64);   // one workgroup per (B,C) plane
    dim3 block(256);      // 8 wave32 waves
    adapool2d_tdm_kernel<<<grid, block, 0, stream>>>(x, beta, out);
}